// MultigridLayer_34239479284132
// MI455X (gfx1250) — compile-verified
//
#include <hip/hip_runtime.h>

typedef float v2f __attribute__((ext_vector_type(2)));
typedef float v8f __attribute__((ext_vector_type(8)));

#define BATCH 32
#define GM 2048
#define GN 1536

// ---------------------------------------------------------------------------
// SYRK: AtA[b] = A[b]^T * A[b], lower-triangular 128x128 blocks only.
// Fused epilogue writes AtA, L, U for the lower+diag region.
// ---------------------------------------------------------------------------
__global__ __launch_bounds__(256) void syrk_wmma_kernel(
    const float* __restrict__ A, float* __restrict__ AtA,
    float* __restrict__ L, float* __restrict__ U) {
  // K-pair swizzled LDS: As2[k>>1][col] = (A[m0+2q][col], A[m0+2q+1][col]).
  // Row stride 144 float2 = 288 floats -> +32 bank shift per row, so the two
  // lane halves (which read adjacent rows) use complementary bank sets.
  __shared__ float2 As2[8][144];
  __shared__ float2 Bs2[8][144];

  // Triangular block decode: blockIdx.x in [0,78) -> (bi,bj), bj<=bi, bi<12.
  int t = blockIdx.x, bi = 0;
  while (t > bi) { t -= (bi + 1); ++bi; }
  const int bj = t;
  const int b = blockIdx.y;

  const int tid = threadIdx.x;
  const int lane = tid & 31, w = tid >> 5;
  const int wr = w & 1, wc = w >> 1;          // 2x4 wave grid over 128x128
  const int ibase = wr * 64, jbase = wc * 32; // wave tile = 64x32
  const int half = lane >> 4, l16 = lane & 15;

  const float* Ab = A + (size_t)b * GM * GN;
  const float* gA = Ab + bi * 128;
  const float* gB = Ab + bj * 128;
  const int lrow = tid >> 5;        // 0..7 (row within 16-chunk; +8 for 2nd)
  const int lcol = (tid & 31) * 4;  // 0..124

  v8f acc[4][2];
#pragma unroll
  for (int it = 0; it < 4; ++it)
#pragma unroll
    for (int jt = 0; jt < 2; ++jt)
      acc[it][jt] = (v8f){0.f, 0.f, 0.f, 0.f, 0.f, 0.f, 0.f, 0.f};

  // register prefetch for chunk m0 = 0
  float4 ra0 = *(const float4*)(gA + (size_t)lrow * GN + lcol);
  float4 ra1 = *(const float4*)(gA + (size_t)(lrow + 8) * GN + lcol);
  float4 rb0 = *(const float4*)(gB + (size_t)lrow * GN + lcol);
  float4 rb1 = *(const float4*)(gB + (size_t)(lrow + 8) * GN + lcol);

  for (int m0 = 0; m0 < GM; m0 += 16) {
    // ---- stage chunk into LDS with K-pair swizzle ----
    {
      const int k0 = lrow, k1 = lrow + 8;
      float* p = (float*)&As2[k0 >> 1][lcol] + (k0 & 1);
      p[0] = ra0.x; p[2] = ra0.y; p[4] = ra0.z; p[6] = ra0.w;
      float* q = (float*)&As2[k1 >> 1][lcol] + (k1 & 1);
      q[0] = ra1.x; q[2] = ra1.y; q[4] = ra1.z; q[6] = ra1.w;
      float* pb = (float*)&Bs2[k0 >> 1][lcol] + (k0 & 1);
      pb[0] = rb0.x; pb[2] = rb0.y; pb[4] = rb0.z; pb[6] = rb0.w;
      float* qb = (float*)&Bs2[k1 >> 1][lcol] + (k1 & 1);
      qb[0] = rb1.x; qb[2] = rb1.y; qb[4] = rb1.z; qb[6] = rb1.w;
    }
    __syncthreads();

    // ---- prefetch next chunk (overlaps WMMA compute) ----
    if (m0 + 16 < GM) {
      const float* nA = gA + (size_t)(m0 + 16) * GN;
      const float* nB = gB + (size_t)(m0 + 16) * GN;
      ra0 = *(const float4*)(nA + (size_t)lrow * GN + lcol);
      ra1 = *(const float4*)(nA + (size_t)(lrow + 8) * GN + lcol);
      rb0 = *(const float4*)(nB + (size_t)lrow * GN + lcol);
      rb1 = *(const float4*)(nB + (size_t)(lrow + 8) * GN + lcol);
    }

    // ---- 4 K-steps of 4: 8 WMMAs each ----
#pragma unroll
    for (int kk = 0; kk < 4; ++kk) {
      const int kk2 = kk * 2 + half;  // lane half selects K={0,1} vs K={2,3}
      v2f afr[4], bfr[2];
#pragma unroll
      for (int it = 0; it < 4; ++it)
        afr[it] = *(const v2f*)&As2[kk2][ibase + it * 16 + l16];
#pragma unroll
      for (int jt = 0; jt < 2; ++jt)
        bfr[jt] = *(const v2f*)&Bs2[kk2][jbase + jt * 16 + l16];
#pragma unroll
      for (int it = 0; it < 4; ++it)
#pragma unroll
        for (int jt = 0; jt < 2; ++jt)
          acc[it][jt] = __builtin_amdgcn_wmma_f32_16x16x4_f32(
              false, afr[it], false, bfr[jt], (short)0, acc[it][jt],
              false, false);
    }
    __syncthreads();
  }

  // ---- epilogue: AtA + fused L/U split (lower+diag region) ----
  const size_t bb = (size_t)b * GN * GN;
  const bool diag = (bi == bj);
#pragma unroll
  for (int it = 0; it < 4; ++it) {
#pragma unroll
    for (int jt = 0; jt < 2; ++jt) {
      union { v8f v; float f[8]; } u;
      u.v = acc[it][jt];
      const int gj = bj * 128 + jbase + jt * 16 + l16;
      const int gib = bi * 128 + ibase + it * 16 + half * 8;
#pragma unroll
      for (int v = 0; v < 8; ++v) {
        const int gi = gib + v;
        const size_t off = bb + (size_t)gi * GN + gj;
        const float val = u.f[v];
        AtA[off] = val;
        if (diag) {
          L[off] = (gj <= gi) ? val : 0.f;
          U[off] = (gj > gi) ? val : 0.f;
        } else {  // whole tile strictly lower
          L[off] = val;
          U[off] = 0.f;
        }
      }
    }
  }
}

// ---------------------------------------------------------------------------
// Mirror lower -> upper via LDS-staged 64x64 transpose (coalesced both ways).
// Also writes U (=mirrored value) and L (=0) for the upper region.
// ---------------------------------------------------------------------------
__global__ __launch_bounds__(256) void mirror_kernel(
    const float* __restrict__ src, float* __restrict__ AtA,
    float* __restrict__ L, float* __restrict__ U) {
  const int tr = blockIdx.x, tc = blockIdx.y, b = blockIdx.z;
  if ((tc >> 1) <= (tr >> 1)) return;  // only strictly-upper 128-block region
  __shared__ float S[64][65];
  const size_t bb = (size_t)b * GN * GN;
  const int tid = threadIdx.x;
#pragma unroll
  for (int q = 0; q < 4; ++q) {
    const int idx = tid + q * 256;
    const int r = idx >> 4, c4 = (idx & 15) * 4;
    const float4 v =
        *(const float4*)&src[bb + (size_t)(tc * 64 + r) * GN + tr * 64 + c4];
    S[r][c4 + 0] = v.x; S[r][c4 + 1] = v.y;
    S[r][c4 + 2] = v.z; S[r][c4 + 3] = v.w;
  }
  __syncthreads();
#pragma unroll
  for (int q = 0; q < 4; ++q) {
    const int idx = tid + q * 256;
    const int r = idx >> 4, c4 = (idx & 15) * 4;
    float4 v;
    v.x = S[c4 + 0][r]; v.y = S[c4 + 1][r];
    v.z = S[c4 + 2][r]; v.w = S[c4 + 3][r];
    const size_t off = bb + (size_t)(tr * 64 + r) * GN + tc * 64 + c4;
    *(float4*)&AtA[off] = v;
    *(float4*)&U[off] = v;
    float4 z; z.x = 0.f; z.y = 0.f; z.z = 0.f; z.w = 0.f;
    *(float4*)&L[off] = z;
  }
}

// ---------------------------------------------------------------------------
// AtPrhs[b,n] = sum_m A[b,m,n]*rhs[b,m];  D[b,n] = sum_m A[b,m,n]^2
// ---------------------------------------------------------------------------
__global__ __launch_bounds__(128) void atprhs_kernel(
    const float* __restrict__ A, const float* __restrict__ Arhs,
    float* __restrict__ AtPrhs, float* __restrict__ D) {
  const int b = blockIdx.y;
  const int col = blockIdx.x * 128 + threadIdx.x;
  const float* Ab = A + (size_t)b * GM * GN + col;
  const float* rb = Arhs + (size_t)b * GM;
  float s1 = 0.f, s2 = 0.f;
#pragma unroll 8
  for (int m = 0; m < GM; ++m) {
    const float a = Ab[(size_t)m * GN];
    const float r = rb[m];
    s1 = fmaf(a, r, s1);
    s2 = fmaf(a, a, s2);
  }
  AtPrhs[b * GN + col] = s1;
  D[b * GN + col] = s2;
}

// ---------------------------------------------------------------------------
// Bilinear (align_corners) downsample of coeffs & rhs + step pair-sums.
// One thread per output element over the 58304 small outputs.
// ---------------------------------------------------------------------------
__device__ __forceinline__ float bilin32(const float* base, int stride_elems,
                                         int yo, int xo) {
  const float scale = 31.0f / 15.0f;
  const float py = yo * scale, px = xo * scale;
  int ly = (int)py; if (ly > 30) ly = 30;
  int lx = (int)px; if (lx > 30) lx = 30;
  const float wy = py - (float)ly, wx = px - (float)lx;
  const float v00 = base[(ly * 32 + lx) * stride_elems];
  const float v01 = base[(ly * 32 + lx + 1) * stride_elems];
  const float v10 = base[((ly + 1) * 32 + lx) * stride_elems];
  const float v11 = base[((ly + 1) * 32 + lx + 1) * stride_elems];
  const float top = v00 * (1.f - wx) + v01 * wx;
  const float bot = v10 * (1.f - wx) + v11 * wx;
  return top * (1.f - wy) + bot * wy;
}

__global__ __launch_bounds__(256) void small_kernel(
    const float* __restrict__ coeffs, const float* __restrict__ rhs,
    const float* __restrict__ sx, const float* __restrict__ sy,
    float* __restrict__ out) {
  const int idx = blockIdx.x * 256 + threadIdx.x;
  if (idx < 49152) {  // new_coeffs [B,256,6]
    const int b = idx / 1536, rem = idx % 1536;
    const int g = rem / 6, o = rem % 6;
    out[idx] = bilin32(coeffs + (size_t)b * 1024 * 6 + o, 6, g >> 4, g & 15);
  } else if (idx < 57344) {  // new_rhs [B,256]
    const int j = idx - 49152;
    const int b = j >> 8, g = j & 255;
    out[idx] = bilin32(rhs + (size_t)b * 1024, 1, g >> 4, g & 15);
  } else if (idx < 57824) {  // nsx [B,15]
    const int j = idx - 57344, b = j / 15, i = j % 15;
    out[idx] = sx[b * 31 + 2 * i] + sx[b * 31 + 2 * i + 1];
  } else if (idx < 58304) {  // nsy [B,15]
    const int j = idx - 57824, b = j / 15, i = j % 15;
    out[idx] = sy[b * 31 + 2 * i] + sy[b * 31 + 2 * i + 1];
  }
}

extern "C" void kernel_launch(void* const* d_in, const int* in_sizes, int n_in,
                              void* d_out, int out_size, void* d_ws,
                              size_t ws_size, hipStream_t stream) {
  const float* coeffs = (const float*)d_in[0];
  const float* rhs = (const float*)d_in[1];
  const float* sx = (const float*)d_in[2];
  const float* sy = (const float*)d_in[3];
  const float* A = (const float*)d_in[4];
  const float* Arhs = (const float*)d_in[5];
  float* out = (float*)d_out;

  const size_t SQ = (size_t)BATCH * GN * GN;  // 75,497,472
  const size_t o_AtA = 58304;
  const size_t o_AtPrhs = o_AtA + SQ;
  const size_t o_D = o_AtPrhs + (size_t)BATCH * GN;
  const size_t o_L = o_D + (size_t)BATCH * GN;
  const size_t o_U = o_L + SQ;

  small_kernel<<<228, 256, 0, stream>>>(coeffs, rhs, sx, sy, out);
  atprhs_kernel<<<dim3(12, BATCH), 128, 0, stream>>>(A, Arhs, out + o_AtPrhs,
                                                     out + o_D);
  syrk_wmma_kernel<<<dim3(78, BATCH), 256, 0, stream>>>(A, out + o_AtA,
                                                        out + o_L, out + o_U);
  mirror_kernel<<<dim3(24, 24, BATCH), 256, 0, stream>>>(
      out + o_AtA, out + o_AtA, out + o_L, out + o_U);
}